// DISConv_43937515438593
// MI455X (gfx1250) — compile-verified
//
#include <hip/hip_runtime.h>
#include <stdint.h>

#define DDIM     64
#define RREL     3
#define OSTRIDE  ((RREL + 1) * DDIM)   // 256 floats per output row
#define BLOCK    256

// ---- CDNA5 async-to-LDS (probe-confirmed present on this toolchain) -------
#if __has_builtin(__builtin_amdgcn_global_load_async_to_lds_b32)
#define HAVE_ASYNC_LDS 1
#else
#define HAVE_ASYNC_LDS 0
#endif

__device__ __forceinline__ float readlane_f32(float v, int lane) {
    return __int_as_float(__builtin_amdgcn_readlane(__float_as_int(v), lane));
}

#if HAVE_ASYNC_LDS
__device__ __forceinline__ void async_copy_b32(const void* g, void* lds) {
    // Builtin params are int* in explicit address spaces (AS1 src, AS3 dst),
    // non-const; 4 args: (gsrc, ldsdst, imm offset, imm cpol).
    // Generic->AS3 via 32-bit truncation (low 32 bits of a generic LDS
    // pointer are the DS offset).
    __builtin_amdgcn_global_load_async_to_lds_b32(
        (__attribute__((address_space(1))) int*)(uintptr_t)g,
        (__attribute__((address_space(3))) int*)(uint32_t)(uintptr_t)lds,
        0, 0);
}
__device__ __forceinline__ void wait_async0() {
#if __has_builtin(__builtin_amdgcn_s_wait_asynccnt)
    __builtin_amdgcn_s_wait_asynccnt(0);
#else
    asm volatile("s_wait_asynccnt 0x0" ::: "memory");
#endif
}
#endif

// ---------------------------------------------------------------------------
// Edge kernel (defined FIRST so the asm snippet shows it).
// One wave handles 32 edges: metadata staged via async->LDS (ASYNCcnt path),
// then per-edge wave-wide row processing: v_readlane broadcasts make
// row/col/val wave-uniform (SGPR), so the x-row gather is saddr-form
// global_load_b64 (float2/lane) and the scatter is two non-returning
// global_atomic_add_f32 per lane into the L2 atomic units.
// ---------------------------------------------------------------------------
__global__ __launch_bounds__(BLOCK) void disconv_edges(const float* __restrict__ x,
                                                       const int*   __restrict__ erows,
                                                       const int*   __restrict__ ecols,
                                                       const float* __restrict__ evals,
                                                       float* __restrict__ out,
                                                       int e_per_rel,
                                                       long long e_total) {
    const int lane      = threadIdx.x & 31;
    const int waveInBlk = threadIdx.x >> 5;
    const long long blockBase = (long long)blockIdx.x * BLOCK;
    const long long base      = blockBase + (long long)waveInBlk * 32;

    // E per relation is a multiple of 32 -> a 32-edge chunk never straddles
    // a relation boundary; rel is wave-uniform.
    long long baseCl = base;
    if (baseCl >= e_total) baseCl = e_total - 32;      // overshoot waves: do 0-adds
    const int rel = (int)(baseCl / (long long)e_per_rel);

    const long long e = base + lane;
    int   r, c;
    float v;

#if HAVE_ASYNC_LDS
    __shared__ int   s_rows[BLOCK];
    __shared__ int   s_cols[BLOCK];
    __shared__ float s_vals[BLOCK];
    {
        long long ge = blockBase + threadIdx.x;
        if (ge >= e_total) ge = e_total - 1;           // keep addresses valid
        async_copy_b32(erows + ge, &s_rows[threadIdx.x]);
        async_copy_b32(ecols + ge, &s_cols[threadIdx.x]);
        async_copy_b32(evals + ge, &s_vals[threadIdx.x]);
    }
    wait_async0();   // per-wave ASYNCcnt; each wave reads only its own slice -> no barrier
    r = s_rows[threadIdx.x];
    c = s_cols[threadIdx.x];
    v = (e < e_total) ? s_vals[threadIdx.x] : 0.0f;    // 0-val => adds 0.0, harmless
#else
    r = 0; c = 0; v = 0.0f;
    if (e < e_total) {
        r = __builtin_nontemporal_load(erows + e);
        c = __builtin_nontemporal_load(ecols + e);
        v = __builtin_nontemporal_load(evals + e);
    }
#endif

    // Pre-touch all 32 source rows (2 x 128B lines each) -> global_prefetch_b8.
    __builtin_prefetch(x + (size_t)c * DDIM, 0, 3);
    __builtin_prefetch(x + (size_t)c * DDIM + 32, 0, 3);

    const float* xbase = x + lane * 2;
    float*       obase = out + rel * DDIM + lane * 2;

#pragma unroll
    for (int i = 0; i < 32; ++i) {
        const int   ri = __builtin_amdgcn_readlane(r, i);   // uniform -> SGPR
        const int   ci = __builtin_amdgcn_readlane(c, i);   // uniform -> SGPR
        const float vi = readlane_f32(v, i);                // uniform -> SGPR

        const float2 xv = *reinterpret_cast<const float2*>(xbase + (size_t)ci * DDIM);
        float* dst = obase + (size_t)ri * OSTRIDE;
        // Non-returning hardware f32 atomic add at the L2 atomic units.
        unsafeAtomicAdd(dst,     xv.x * vi);
        unsafeAtomicAdd(dst + 1, xv.y * vi);
    }
}

// ---------------------------------------------------------------------------
// Init kernel: out[:, 0:192] = 0 ; out[:, 192:256] = x. One float4 per thread.
// Runs every call (harness poisons d_out; replays accumulate on top of init).
// ---------------------------------------------------------------------------
__global__ __launch_bounds__(BLOCK) void disconv_init(const float* __restrict__ x,
                                                      float* __restrict__ out,
                                                      int n) {
    int idx   = blockIdx.x * blockDim.x + threadIdx.x;   // over n * 64 float4s
    int total = n * (OSTRIDE / 4);
    if (idx >= total) return;
    int row = idx >> 6;
    int c4  = idx & 63;
    float4 v;
    if (c4 < (RREL * DDIM) / 4) {
        v = make_float4(0.f, 0.f, 0.f, 0.f);
    } else {
        const float4* x4 = reinterpret_cast<const float4*>(x);
        v = x4[row * (DDIM / 4) + (c4 - (RREL * DDIM) / 4)];
    }
    reinterpret_cast<float4*>(out)[idx] = v;
}

extern "C" void kernel_launch(void* const* d_in, const int* in_sizes, int n_in,
                              void* d_out, int out_size, void* d_ws, size_t ws_size,
                              hipStream_t stream) {
    const float* x     = (const float*)d_in[0];
    const int*   erows = (const int*)  d_in[1];
    const int*   ecols = (const int*)  d_in[2];
    const float* evals = (const float*)d_in[3];
    float* out = (float*)d_out;

    const int       n     = in_sizes[0] / DDIM;        // 50000
    const long long e_tot = (long long)in_sizes[1];    // R * E = 2.4M
    const int       e_rel = (int)(e_tot / RREL);       // 800000

    {
        int total4 = n * (OSTRIDE / 4);
        int blocks = (total4 + BLOCK - 1) / BLOCK;
        hipLaunchKernelGGL(disconv_init, dim3(blocks), dim3(BLOCK), 0, stream, x, out, n);
    }
    {
        long long waves  = (e_tot + 31) / 32;
        long long blocks = (waves + 7) / 8;
        hipLaunchKernelGGL(disconv_edges, dim3((unsigned)blocks), dim3(BLOCK), 0, stream,
                           x, erows, ecols, evals, out, e_rel, e_tot);
    }
}